// ATSSAssigner_51445118272108
// MI455X (gfx1250) — compile-verified
//
#include <hip/hip_runtime.h>
#include <stdint.h>

// ---------------- problem constants (from reference) ----------------
#define TOPK   9
#define NCLS   80
#define BATCH  16
#define MGT    64
#define NA     33600          // 25600 + 6400 + 1600
#define EPSF   1e-9f

typedef __attribute__((ext_vector_type(2))) float v2f;
typedef __attribute__((ext_vector_type(8))) float v8f;

__device__ __forceinline__ float iou_fn(float gx0, float gy0, float gx1, float gy1, float garea,
                                        float ax0, float ay0, float ax1, float ay1, float aarea) {
    float ix0 = fmaxf(gx0, ax0), iy0 = fmaxf(gy0, ay0);
    float ix1 = fminf(gx1, ax1), iy1 = fminf(gy1, ay1);
    float iw = fmaxf(ix1 - ix0, 0.f), ih = fmaxf(iy1 - iy0, 0.f);
    float ov = iw * ih;
    return ov / (garea + aarea - ov + EPSF);
}

// Pack (distance, index) into one totally-ordered u64: smaller distance first,
// ties broken by smaller anchor index (matches jax.lax.top_k ordering).
__device__ __forceinline__ uint64_t pack_key(float d2, int idx) {
    uint32_t u = __float_as_uint(d2);
    u = (u & 0x80000000u) ? ~u : (u | 0x80000000u);   // total-order float transform
    return ((uint64_t)u << 32) | (uint32_t)idx;
}

// Branch-free sorted insert into a 9-deep register list (ascending):
// lowers to a v_min_u64/v_max_u64 chain, no control flow.
__device__ __forceinline__ void insert9(uint64_t (&lst)[TOPK], uint64_t kv) {
    #pragma unroll
    for (int k = 0; k < TOPK; ++k) {
        uint64_t cur = lst[k];
        bool sw = kv < cur;
        lst[k] = sw ? kv : cur;
        kv     = sw ? cur : kv;
    }
}

// ---------------- workspace init ----------------
__global__ void atss_init(int* __restrict__ mps, int* __restrict__ minm, int n) {
    int t = blockIdx.x * blockDim.x + threadIdx.x;
    if (t < n) { mps[t] = 0; minm[t] = 0x7fffffff; }
}

// ---------------- phase 1: WMMA distance + per-level top-9 + threshold + scatter ----
// grid: B * 4 (one batch, one tile of 16 GTs per block); block: 512 = 16 waves.
// d2[m,n] via V_WMMA_F32_16X16X4_F32 with A[m,:]=[gx,gy,|g|^2,1], B[:,n]=[-2ax,-2ay,1,|a|^2].
__launch_bounds__(512)
__global__ void atss_phase1(const float* __restrict__ anchors,
                            const float* __restrict__ gtb_g,
                            const float* __restrict__ padmask,
                            int* __restrict__ mps, int* __restrict__ minm)
{
    __shared__ float    s_gcx[16], s_gcy[16], s_gg[16], s_area[16], s_thr[16];
    __shared__ float    s_gbox[16][4];
    __shared__ int      s_valid[16];
    __shared__ float    s_d2[16][256];              // per-wave 16x16 d^2 tile
    __shared__ uint64_t s_cand[16 * 32 * TOPK];     // [gt][list][k] packed (d2,idx)
    __shared__ int      s_topk[16][3 * TOPK];
    __shared__ float    s_iou[16][3 * TOPK];
    __shared__ int      s_in[16][3 * TOPK];

    const int b    = blockIdx.x >> 2;
    const int tile = blockIdx.x & 3;
    const int tid  = threadIdx.x;
    const int wave = tid >> 5;
    const int lane = tid & 31;
    const int g    = lane & 15;   // gt owned by this lane for selection
    const int h    = lane >> 4;   // column half

    if (tid < 16) {
        int m = tile * 16 + tid;
        const float4 gb = reinterpret_cast<const float4*>(gtb_g)[b * MGT + m];
        s_gbox[tid][0] = gb.x; s_gbox[tid][1] = gb.y; s_gbox[tid][2] = gb.z; s_gbox[tid][3] = gb.w;
        float cx = (gb.x + gb.z) * 0.5f, cy = (gb.y + gb.w) * 0.5f;
        s_gcx[tid] = cx; s_gcy[tid] = cy; s_gg[tid] = cx * cx + cy * cy;
        s_area[tid] = (gb.z - gb.x) * (gb.w - gb.y);
        s_valid[tid] = (padmask[b * MGT + m] > 0.f) ? 1 : 0;
    }
    __syncthreads();

    // A-matrix operand: constant for all anchor tiles of this block.
    // lanes 0-15: v0=A[m,0]=gx, v1=A[m,1]=gy ; lanes 16-31: v0=A[m,2]=|g|^2, v1=A[m,3]=1
    v2f av;
    if (lane < 16) { av.x = s_gcx[lane];      av.y = s_gcy[lane]; }
    else           { av.x = s_gg[lane - 16];  av.y = 1.0f; }

    const float4* anchors4 = reinterpret_cast<const float4*>(anchors);
    const int lvl_start[3] = {0, 25600, 32000};
    const int lvl_n[3]     = {25600, 6400, 1600};

    for (int lvl = 0; lvl < 3; ++lvl) {
        uint64_t lst[TOPK];
        #pragma unroll
        for (int k = 0; k < TOPK; ++k) lst[k] = ~0ull;

        const int base   = lvl_start[lvl];
        const int ntiles = lvl_n[lvl] >> 4;
        const int nl     = lane & 15;

        // software-pipelined anchor-tile fetch: load t+16 while processing t
        int t = wave;
        float4 ab = make_float4(0.f, 0.f, 0.f, 0.f);
        if (t < ntiles) ab = anchors4[base + (t << 4) + nl];

        while (t < ntiles) {                   // wave-uniform trip condition
            const int tn = t + 16;
            float4 abn = ab;
            if (tn < ntiles) abn = anchors4[base + (tn << 4) + nl];      // in flight
            if (tn + 16 < ntiles)                                         // L0 prefetch
                __builtin_prefetch(&anchors4[base + ((tn + 16) << 4) + nl], 0, 3);

            // B-matrix operand from current tile's anchors
            float acx = (ab.x + ab.z) * 0.5f, acy = (ab.y + ab.w) * 0.5f;
            v2f bv;
            if (lane < 16) { bv.x = -2.f * acx; bv.y = -2.f * acy; }
            else           { bv.x = 1.0f;       bv.y = acx * acx + acy * acy; }

            v8f c = {};
            v8f d = __builtin_amdgcn_wmma_f32_16x16x4_f32(false, av, false, bv,
                                                          (short)0, c, false, false);

            // stage D tile to per-wave LDS: VGPR r -> row (r + 8*half), col = lane&15
            float* tb = &s_d2[wave][0];
            const int col   = lane & 15;
            const int rbase = (lane < 16) ? 0 : 8;
            #pragma unroll
            for (int r = 0; r < 8; ++r) tb[(rbase + r) * 16 + col] = d[r];
            asm volatile("s_wait_dscnt 0x0" ::: "memory"); // same-wave cross-lane handoff

            // lane (g,h) scans columns h*8..h*8+7 of gt row g: two ds_load_b128
            const int bc = h * 8;
            const float4* tb4 = reinterpret_cast<const float4*>(&s_d2[wave][g * 16 + bc]);
            float4 va = tb4[0];
            float4 vb = tb4[1];
            const int di = (t << 4) + bc;
            insert9(lst, pack_key(va.x, di));
            insert9(lst, pack_key(va.y, di + 1));
            insert9(lst, pack_key(va.z, di + 2));
            insert9(lst, pack_key(va.w, di + 3));
            insert9(lst, pack_key(vb.x, di + 4));
            insert9(lst, pack_key(vb.y, di + 5));
            insert9(lst, pack_key(vb.z, di + 6));
            insert9(lst, pack_key(vb.w, di + 7));

            ab = abn;
            t  = tn;
        }

        // dump 32 partial lists per gt, then one thread per gt selects global top-9
        const int list = wave * 2 + h;
        #pragma unroll
        for (int k = 0; k < TOPK; ++k)
            s_cand[(g * 32 + list) * TOPK + k] = lst[k];
        __syncthreads();
        if (tid < 16) {
            uint64_t* ck = &s_cand[tid * 32 * TOPK];
            for (int q = 0; q < TOPK; ++q) {
                uint64_t best = ~0ull; int bs = 0;
                for (int s = 0; s < 32 * TOPK; ++s) {
                    uint64_t v = ck[s];
                    bool lt = v < best;
                    best = lt ? v : best;
                    bs   = lt ? s : bs;
                }
                ck[bs] = ~0ull;
                s_topk[tid][lvl * TOPK + q] = base + (int)(best & 0x7fffffffu);
            }
        }
        __syncthreads();
    }

    // candidate IoUs + inside-gt flags (432 tasks)
    if (tid < 16 * 3 * TOPK) {
        int gg = tid / 27, k = tid % 27;
        int aidx = s_topk[gg][k];
        const float4 ab = reinterpret_cast<const float4*>(anchors)[aidx];
        float aarea = (ab.z - ab.x) * (ab.w - ab.y);
        float gx0 = s_gbox[gg][0], gy0 = s_gbox[gg][1], gx1 = s_gbox[gg][2], gy1 = s_gbox[gg][3];
        s_iou[gg][k] = iou_fn(gx0, gy0, gx1, gy1, s_area[gg], ab.x, ab.y, ab.z, ab.w, aarea);
        float acx = (ab.x + ab.z) * 0.5f, acy = (ab.y + ab.w) * 0.5f;
        float mn = fminf(fminf(acx - gx0, acy - gy0), fminf(gx1 - acx, gy1 - acy));
        s_in[gg][k] = (mn > EPSF) ? 1 : 0;
    }
    __syncthreads();
    if (tid < 16) {   // thr = mean + std(ddof=1) over the 27 candidate IoUs
        float sum = 0.f;
        for (int k = 0; k < 27; ++k) sum += s_iou[tid][k];
        float mean = sum * (1.0f / 27.0f);
        float ss = 0.f;
        for (int k = 0; k < 27; ++k) { float d = s_iou[tid][k] - mean; ss += d * d; }
        s_thr[tid] = mean + sqrtf(ss * (1.0f / 26.0f));
    }
    __syncthreads();
    if (tid < 16 * 3 * TOPK) {   // scatter positives
        int gg = tid / 27, k = tid % 27;
        if (s_valid[gg] && s_in[gg][k] && (s_iou[gg][k] > s_thr[gg])) {
            int aidx = s_topk[gg][k];
            int m = tile * 16 + gg;
            atomicAdd(&mps[b * NA + aidx], 1);
            atomicMin(&minm[b * NA + aidx], m);
        }
    }
}

// ---------------- phase 2: per-anchor resolve (labels + bboxes) ----------------
__global__ void atss_resolve(const float* __restrict__ anchors,
                             const float* __restrict__ gtb_g,
                             const int*   __restrict__ gtlab,
                             const int*   __restrict__ mps,
                             const int*   __restrict__ minm,
                             int*   __restrict__ out_lab,
                             float* __restrict__ out_box)
{
    int t = blockIdx.x * blockDim.x + threadIdx.x;
    if (t >= BATCH * NA) return;
    int b = t / NA, a = t - b * NA;
    int c = mps[t];
    int g = 0, label = NCLS;               // background default; bbox gathers gt[b,0]
    if (c == 1) {
        g = minm[t];                       // argmax of a single-1 column = first (min) m
        label = gtlab[b * MGT + g];
    } else if (c > 1) {                    // multi-claim: argmax IoU over ALL 64 gts (incl. pads)
        const float4 ab = reinterpret_cast<const float4*>(anchors)[a];
        float aarea = (ab.z - ab.x) * (ab.w - ab.y);
        float best = -1.0f; int bm = 0;
        for (int m = 0; m < MGT; ++m) {
            const float4 gb = reinterpret_cast<const float4*>(gtb_g)[b * MGT + m];
            float garea = (gb.z - gb.x) * (gb.w - gb.y);
            float iou = iou_fn(gb.x, gb.y, gb.z, gb.w, garea, ab.x, ab.y, ab.z, ab.w, aarea);
            bool gt = iou > best;          // strict > keeps first occurrence
            best = gt ? iou : best;
            bm   = gt ? m   : bm;
        }
        g = bm; label = gtlab[b * MGT + g];
    }
    out_lab[t] = label;
    const float4 gb = reinterpret_cast<const float4*>(gtb_g)[b * MGT + g];
    reinterpret_cast<float4*>(out_box)[t] = gb;
}

// ---------------- phase 3: one-hot scores, fully coalesced float4 (BW-bound) ------
__global__ void atss_scores(const int* __restrict__ lab, float* __restrict__ out_sc, int total4)
{
    int t = blockIdx.x * blockDim.x + threadIdx.x;
    if (t >= total4) return;
    int a = t / 20, q = t - a * 20;
    int label = lab[a];
    int c0 = q * 4;
    float4 v;
    v.x = (label == c0    ) ? 1.f : 0.f;
    v.y = (label == c0 + 1) ? 1.f : 0.f;
    v.z = (label == c0 + 2) ? 1.f : 0.f;
    v.w = (label == c0 + 3) ? 1.f : 0.f;
    reinterpret_cast<float4*>(out_sc)[t] = v;
}

extern "C" void kernel_launch(void* const* d_in, const int* in_sizes, int n_in,
                              void* d_out, int out_size, void* d_ws, size_t ws_size,
                              hipStream_t stream)
{
    const float* anchors = (const float*)d_in[0];   // [A,4] f32
    // d_in[1] = num_anchors_list (compile-time constants), d_in[5] = bg_index (=80)
    const int*   gtlab   = (const int*)d_in[2];     // [B,M,1] i32
    const float* gtb     = (const float*)d_in[3];   // [B,M,4] f32
    const float* pmask   = (const float*)d_in[4];   // [B,M,1] f32

    int* mps  = (int*)d_ws;                 // [B*A]
    int* minm = mps + BATCH * NA;           // [B*A]

    int*   out_lab = (int*)d_out;                                   // [B*A]   i32 bits
    float* out_box = (float*)d_out + BATCH * NA;                    // [B*A,4] f32
    float* out_sc  = (float*)d_out + BATCH * NA + BATCH * NA * 4;   // [B*A,80] f32

    const int n = BATCH * NA;
    atss_init   <<<(n + 255) / 256, 256, 0, stream>>>(mps, minm, n);
    atss_phase1 <<<BATCH * 4, 512, 0, stream>>>(anchors, gtb, pmask, mps, minm);
    atss_resolve<<<(n + 255) / 256, 256, 0, stream>>>(anchors, gtb, gtlab, mps, minm,
                                                      out_lab, out_box);
    const int t4 = n * 20;
    atss_scores <<<(t4 + 255) / 256, 256, 0, stream>>>(out_lab, out_sc, t4);
}